// SoftmaxLoss_28054726377853
// MI455X (gfx1250) — compile-verified
//
#include <hip/hip_runtime.h>
#include <hip/hip_bf16.h>
#include <math.h>

// Problem constants from the reference
#define N_SPK   128
#define M_UTT   16
#define P_DIM   256
#define K_ROWS  (N_SPK * M_UTT)   // 2048
#define W_SCALE 10.0f
#define B_BIAS  (-5.0f)
#define EPS_NORM 1e-12f
#define EPS_LOG  1e-9f

typedef __attribute__((ext_vector_type(2))) float v2f;
typedef __attribute__((ext_vector_type(8))) float v8f;

// ---------------- workspace layout (in floats) ----------------
#define WS_CENT  0          // 128*256 raw centroid sums
#define WS_CN    32768      // 128*256 normalized centroids
#define WS_EN    65536      // 2048*256 normalized e rows
#define WS_SDIAG 589824     // 2048 diagonal similarities (already 10x-5)
#define WS_TERMS 591872     // 2048 per-row loss terms

// ---------------------------------------------------------------
// Kernel 1: centroid sums over M utterances, and normalized Cn.
// grid = 128 (one block per speaker n), block = 256 (one thread per p)
__global__ __launch_bounds__(256) void k_centroids(const float* __restrict__ e,
                                                   float* __restrict__ cent,
                                                   float* __restrict__ cn) {
    const int n = blockIdx.x;
    const int p = threadIdx.x;

    float s = 0.0f;
#pragma unroll
    for (int m = 0; m < M_UTT; ++m)
        s += e[(n * M_UTT + m) * P_DIM + p];

    cent[n * P_DIM + p] = s;

    // off-diagonal centroid = cent_sum / M, then l2-normalize
    float c = s * (1.0f / (float)M_UTT);

    __shared__ float red[256];
    red[p] = c * c;
    __syncthreads();
    for (int off = 128; off > 0; off >>= 1) {
        if (p < off) red[p] += red[p + off];
        __syncthreads();
    }
    float inv = 1.0f / sqrtf(red[0] + EPS_NORM);
    cn[n * P_DIM + p] = c * inv;
}

// ---------------------------------------------------------------
// Kernel 2: per-row normalized e, plus diagonal similarity term.
// grid = 2048 (one block per row k), block = 256 (one thread per p)
__global__ __launch_bounds__(256) void k_rows(const float* __restrict__ e,
                                              const float* __restrict__ cent,
                                              float* __restrict__ en,
                                              float* __restrict__ sdiag) {
    const int k = blockIdx.x;
    const int p = threadIdx.x;
    const int n = k >> 4;   // k / M, M == 16

    const float ev = e[k * P_DIM + p];
    const float cs = cent[n * P_DIM + p];
    const float cd = (cs - ev) * (1.0f / (float)(M_UTT - 1));

    __shared__ float r0[256], r1[256], r2[256];
    r0[p] = ev * ev;
    r1[p] = cd * cd;
    r2[p] = ev * cd;
    __syncthreads();
    for (int off = 128; off > 0; off >>= 1) {
        if (p < off) {
            r0[p] += r0[p + off];
            r1[p] += r1[p + off];
            r2[p] += r2[p + off];
        }
        __syncthreads();
    }
    const float see = r0[0], scc = r1[0], sec = r2[0];

    en[k * P_DIM + p] = ev / sqrtf(see + EPS_NORM);

    if (p == 0) {
        float cosd = sec / (sqrtf(see + EPS_NORM) * sqrtf(scc + EPS_NORM));
        sdiag[k] = W_SCALE * cosd + B_BIAS;
    }
}

// ---------------------------------------------------------------
// Kernel 3: S = En(2048x256) @ Cn^T via V_WMMA_F32_16X16X4_F32.
// grid = 128 row-tiles, block = 256 threads = 8 waves; wave w owns the
// 16x16 tile at (rowBase = 16*blockIdx.x, colBase = 16*w). K-loop = 64.
// A-operand layout (16x4 f32): lanes 0-15 hold K=0 (vgpr0), K=1 (vgpr1);
// lanes 16-31 hold K=2, K=3. B (4x16) mirrors this with N across lanes.
// Epilogue is fully uniform (no divergence): diagonal entries get patched
// later in k_lse, which owns them anyway.
__global__ __launch_bounds__(256) void k_gemm_wmma(const float* __restrict__ en,
                                                   const float* __restrict__ cn,
                                                   float* __restrict__ S) {
    const int wave = threadIdx.x >> 5;
    const int lane = threadIdx.x & 31;
    const int half = lane >> 4;     // 0: K=0,1   1: K=2,3
    const int l16  = lane & 15;

    const int rowBase = blockIdx.x * 16;
    const int colBase = wave * 16;

    const float* aptr = en + (size_t)(rowBase + l16) * P_DIM + 2 * half;
    const float* bptr = cn + (size_t)(colBase + l16) * P_DIM + 2 * half;

    __builtin_prefetch(aptr, 0, 1);          // global_prefetch_b8
    __builtin_prefetch(bptr, 0, 1);

    v8f c = {};
#pragma unroll 8
    for (int kk = 0; kk < P_DIM / 4; ++kk) {
        v2f a = *(const v2f*)(aptr + kk * 4);
        v2f b = *(const v2f*)(bptr + kk * 4);
        c = __builtin_amdgcn_wmma_f32_16x16x4_f32(
                /*neg_a=*/false, a, /*neg_b=*/false, b,
                /*c_mod=*/(short)0, c, /*reuse_a=*/false, /*reuse_b=*/false);
    }

    // Branch-free epilogue: s = 10*s - 5, store.
    const int n = colBase + l16;
    float* outp = S + (size_t)rowBase * N_SPK + n;
#pragma unroll
    for (int j = 0; j < 8; ++j) {
        const int m = j + 8 * half;          // C/D layout: M = vgpr + 8*(lane>=16)
        outp[(size_t)m * N_SPK] = W_SCALE * c[j] + B_BIAS;
    }
}

// ---------------------------------------------------------------
// Kernel 4: patch the diagonal entry of each row with sdiag[k], then
// compute the per-row loss term. One wave32 per row; 8 waves per block.
// grid = 256 blocks, block = 256 threads.
__global__ __launch_bounds__(256) void k_lse(float* __restrict__ S,
                                             const float* __restrict__ sdiag,
                                             float* __restrict__ terms) {
    const int wave = threadIdx.x >> 5;
    const int lane = threadIdx.x & 31;
    const int k = blockIdx.x * 8 + wave;

    float* row = S + (size_t)k * N_SPK;
    const int   diag = k >> 4;      // diagonal column for row k
    const float sd   = sdiag[k];    // correct diagonal value (already 10x-5)

    float sum = 0.0f;
#pragma unroll
    for (int j = 0; j < N_SPK / 32; ++j) {
        const int idx = lane + 32 * j;
        float v = row[idx];
        if (idx == diag) {          // exactly one lane/iteration per row
            v = sd;
            row[idx] = sd;          // publish corrected diagonal to output s
        }
        sum += expf(v);
    }

    // wave32 butterfly reduction
#pragma unroll
    for (int off = 16; off > 0; off >>= 1)
        sum += __shfl_xor(sum, off, 32);

    if (lane == 0)
        terms[k] = -sd + logf(sum + EPS_LOG);
}

// ---------------------------------------------------------------
// Kernel 5: deterministic final reduction of 2048 terms -> loss.
// single block of 256 threads, fixed-order tree.
__global__ __launch_bounds__(256) void k_loss(const float* __restrict__ terms,
                                              float* __restrict__ loss) {
    const int t = threadIdx.x;
    float s = 0.0f;
#pragma unroll
    for (int j = 0; j < K_ROWS / 256; ++j)
        s += terms[t + 256 * j];

    __shared__ float red[256];
    red[t] = s;
    __syncthreads();
    for (int off = 128; off > 0; off >>= 1) {
        if (t < off) red[t] += red[t + off];
        __syncthreads();
    }
    if (t == 0) loss[0] = red[0] * (1.0f / (float)K_ROWS);
}

// ---------------------------------------------------------------
extern "C" void kernel_launch(void* const* d_in, const int* in_sizes, int n_in,
                              void* d_out, int out_size, void* d_ws, size_t ws_size,
                              hipStream_t stream) {
    (void)in_sizes; (void)n_in; (void)out_size; (void)ws_size;

    const float* e = (const float*)d_in[0];
    float* out = (float*)d_out;              // out[0] = loss, out[1..] = s

    float* ws    = (float*)d_ws;
    float* cent  = ws + WS_CENT;
    float* cn    = ws + WS_CN;
    float* en    = ws + WS_EN;
    float* sdiag = ws + WS_SDIAG;
    float* terms = ws + WS_TERMS;

    float* S = out + 1;

    k_centroids<<<N_SPK,       256, 0, stream>>>(e, cent, cn);
    k_rows     <<<K_ROWS,      256, 0, stream>>>(e, cent, en, sdiag);
    k_gemm_wmma<<<K_ROWS / 16, 256, 0, stream>>>(en, cn, S);
    k_lse      <<<K_ROWS / 8,  256, 0, stream>>>(S, sdiag, terms);
    k_loss     <<<1,           256, 0, stream>>>(terms, out);
}